// SparseMoEFFN_36043365548776
// MI455X (gfx1250) — compile-verified
//
#include <hip/hip_runtime.h>
#include <math.h>

#define NTOK 2048
#define DM   2048
#define DF   4096
#define NEXP 8
#define BM   128
#define BN   128
#define BK   16
#define MCAP 5120          // 40 M-blocks; >= 4096 slots + 8*127 padding
#define MBLK (MCAP / BM)
#define LDA  20            // padded LDS stride for A (mult of 4, conflict-free)

typedef float v2f __attribute__((ext_vector_type(2)));
typedef float v8f __attribute__((ext_vector_type(8)));

// ---------------- Router: logits -> softmax -> top2 -> weights ----------------
__global__ __launch_bounds__(256) void moe_router(const float* __restrict__ x,
                                                  const float* __restrict__ Wr,
                                                  int* __restrict__ tk_i,
                                                  float* __restrict__ tk_w,
                                                  int* __restrict__ counts,
                                                  float* __restrict__ psum) {
  __shared__ float red[256];
  __shared__ float logits[NEXP];
  const int n = blockIdx.x;
  const int tid = threadIdx.x;
  const float* xr = x + (size_t)n * DM;
  float acc[NEXP];
#pragma unroll
  for (int e = 0; e < NEXP; ++e) acc[e] = 0.0f;
  for (int d = tid; d < DM; d += 256) {
    float xv = xr[d];
#pragma unroll
    for (int e = 0; e < NEXP; ++e) acc[e] += xv * Wr[e * DM + d];
  }
  for (int e = 0; e < NEXP; ++e) {
    red[tid] = acc[e];
    __syncthreads();
    for (int s = 128; s > 0; s >>= 1) {
      if (tid < s) red[tid] += red[tid + s];
      __syncthreads();
    }
    if (tid == 0) logits[e] = red[0];
    __syncthreads();
  }
  if (tid == 0) {
    float m = logits[0];
#pragma unroll
    for (int e = 1; e < NEXP; ++e) m = fmaxf(m, logits[e]);
    float p[NEXP], s = 0.0f;
#pragma unroll
    for (int e = 0; e < NEXP; ++e) { p[e] = expf(logits[e] - m); s += p[e]; }
#pragma unroll
    for (int e = 0; e < NEXP; ++e) p[e] /= s;
    int i0 = 0;
#pragma unroll
    for (int e = 1; e < NEXP; ++e) if (p[e] > p[i0]) i0 = e;
    int i1 = (i0 == 0) ? 1 : 0;
#pragma unroll
    for (int e = 0; e < NEXP; ++e) if (e != i0 && p[e] > p[i1]) i1 = e;
    float s2 = p[i0] + p[i1];
    tk_i[n * 2 + 0] = i0;
    tk_i[n * 2 + 1] = i1;
    tk_w[n * 2 + 0] = p[i0] / s2;
    tk_w[n * 2 + 1] = p[i1] / s2;
    atomicAdd(&counts[i0], 1);
    atomicAdd(&counts[i1], 1);
#pragma unroll
    for (int e = 0; e < NEXP; ++e) atomicAdd(&psum[e], p[e]);
  }
}

// ------------- Prefix: padded per-expert regions, block->expert, aux ----------
__global__ void moe_prefix(const int* __restrict__ counts,
                           const float* __restrict__ psum,
                           int* __restrict__ cursors,
                           int* __restrict__ block_expert,
                           float* __restrict__ out_aux) {
  if (threadIdx.x == 0 && blockIdx.x == 0) {
    int start = 0;
    for (int e = 0; e < NEXP; ++e) {
      cursors[e] = start;
      int nb = (counts[e] + BM - 1) / BM;
      for (int b = 0; b < nb; ++b) block_expert[start / BM + b] = e;
      start += nb * BM;
    }
    for (int b = start / BM; b < MBLK; ++b) block_expert[b] = 0;
    float aux = 0.0f;
    for (int e = 0; e < NEXP; ++e) {
      float fi = (float)counts[e] / (float)(NTOK * 2);
      float Pi = psum[e] / (float)NTOK;
      aux += fi * Pi;
    }
    out_aux[0] = 0.01f * (float)NEXP * aux;
  }
}

// -------------------- Scatter tokens into per-expert slots --------------------
__global__ __launch_bounds__(256) void moe_scatter(const int* __restrict__ tk_i,
                                                   const float* __restrict__ tk_w,
                                                   int* __restrict__ cursors,
                                                   int* __restrict__ slot_token,
                                                   float* __restrict__ slot_w) {
  int n = blockIdx.x * blockDim.x + threadIdx.x;
  if (n < NTOK) {
#pragma unroll
    for (int k = 0; k < 2; ++k) {
      int e = tk_i[n * 2 + k];
      int pos = atomicAdd(&cursors[e], 1);
      slot_token[pos] = n;
      slot_w[pos] = tk_w[n * 2 + k];
    }
  }
}

// ---------- GEMM1: h = silu(X W1) * (X W3), gathered rows, fp32 WMMA ----------
__global__ __launch_bounds__(256) void moe_gemm1(const float* __restrict__ x,
                                                 const float* __restrict__ W1,
                                                 const float* __restrict__ W3,
                                                 const int* __restrict__ slot_token,
                                                 const int* __restrict__ block_expert,
                                                 float* __restrict__ h) {
  __shared__ float lA[BM * LDA];
  __shared__ float lB1[BK * BN];
  __shared__ float lB3[BK * BN];
  const int tid = threadIdx.x;
  const int bm = blockIdx.y, bn = blockIdx.x;
  const int e = block_expert[bm];
  const float* W1e = W1 + (size_t)e * DM * DF;
  const float* W3e = W3 + (size_t)e * DM * DF;
  const int fb = bn * BN;

  const int arow0 = tid >> 2;            // 0..63 (second row = +64)
  const int ac4 = (tid & 3) * 4;         // 0,4,8,12 within BK
  const int tok0 = slot_token[bm * BM + arow0];
  const int tok1 = slot_token[bm * BM + arow0 + 64];
  const float* ap0 = x + (size_t)(tok0 >= 0 ? tok0 : 0) * DM;
  const float* ap1 = x + (size_t)(tok1 >= 0 ? tok1 : 0) * DM;

  const int brow0 = tid >> 5;            // 0..7 (second row = +8)
  const int bc4 = (tid & 31) * 4;

  const int wid = tid >> 5;
  const int lane = tid & 31;
  const int wm = wid & 3;                // 4 waves over M: 32 rows each
  const int wn = wid >> 2;               // 2 waves over N: 64 cols each
  const int l16 = lane & 15;
  const int lhi = lane >> 4;

  v8f accg[2][4], accu[2][4];
#pragma unroll
  for (int i = 0; i < 2; ++i)
#pragma unroll
    for (int j = 0; j < 4; ++j) { accg[i][j] = (v8f)0.0f; accu[i][j] = (v8f)0.0f; }

  for (int k0 = 0; k0 < DM; k0 += BK) {
    float4 va0 = make_float4(0.f, 0.f, 0.f, 0.f);
    float4 va1 = make_float4(0.f, 0.f, 0.f, 0.f);
    if (tok0 >= 0) va0 = *(const float4*)(ap0 + k0 + ac4);
    if (tok1 >= 0) va1 = *(const float4*)(ap1 + k0 + ac4);
    *(float4*)&lA[arow0 * LDA + ac4] = va0;
    *(float4*)&lA[(arow0 + 64) * LDA + ac4] = va1;
#pragma unroll
    for (int i = 0; i < 2; ++i) {
      int br = brow0 + i * 8;
      *(float4*)&lB1[br * BN + bc4] = *(const float4*)&W1e[(size_t)(k0 + br) * DF + fb + bc4];
      *(float4*)&lB3[br * BN + bc4] = *(const float4*)&W3e[(size_t)(k0 + br) * DF + fb + bc4];
    }
    __syncthreads();
#pragma unroll
    for (int kk = 0; kk < BK; kk += 4) {
      const int kin = kk + lhi * 2;
      v2f af[2], b1f[4], b3f[4];
#pragma unroll
      for (int tm = 0; tm < 2; ++tm) {
        int r = wm * 32 + tm * 16 + l16;
        af[tm].x = lA[r * LDA + kin];
        af[tm].y = lA[r * LDA + kin + 1];
      }
#pragma unroll
      for (int tn = 0; tn < 4; ++tn) {
        int c = wn * 64 + tn * 16 + l16;
        b1f[tn].x = lB1[kin * BN + c];
        b1f[tn].y = lB1[(kin + 1) * BN + c];
        b3f[tn].x = lB3[kin * BN + c];
        b3f[tn].y = lB3[(kin + 1) * BN + c];
      }
#pragma unroll
      for (int tm = 0; tm < 2; ++tm)
#pragma unroll
        for (int tn = 0; tn < 4; ++tn) {
          accg[tm][tn] = __builtin_amdgcn_wmma_f32_16x16x4_f32(
              false, af[tm], false, b1f[tn], (short)0, accg[tm][tn], false, false);
          accu[tm][tn] = __builtin_amdgcn_wmma_f32_16x16x4_f32(
              false, af[tm], false, b3f[tn], (short)0, accu[tm][tn], false, false);
        }
    }
    __syncthreads();
  }
#pragma unroll
  for (int tm = 0; tm < 2; ++tm)
#pragma unroll
    for (int tn = 0; tn < 4; ++tn) {
      int rbase = bm * BM + wm * 32 + tm * 16 + lhi * 8;
      int cbase = fb + wn * 64 + tn * 16 + l16;
#pragma unroll
      for (int v = 0; v < 8; ++v) {
        float g = accg[tm][tn][v];
        float u = accu[tm][tn][v];
        float sg = g / (1.0f + expf(-g));
        h[(size_t)(rbase + v) * DF + cbase] = sg * u;
      }
    }
}

// ------------- GEMM2: out[token] += w * (h @ W2[e]), fp32 WMMA ---------------
__global__ __launch_bounds__(256) void moe_gemm2(const float* __restrict__ h,
                                                 const float* __restrict__ W2,
                                                 const int* __restrict__ slot_token,
                                                 const float* __restrict__ slot_w,
                                                 const int* __restrict__ block_expert,
                                                 float* __restrict__ out) {
  __shared__ float lA[BM * LDA];
  __shared__ float lB[BK * BN];
  const int tid = threadIdx.x;
  const int bm = blockIdx.y, bn = blockIdx.x;
  const int e = block_expert[bm];
  const float* W2e = W2 + (size_t)e * DF * DM;
  const int db = bn * BN;

  const int arow0 = tid >> 2;
  const int ac4 = (tid & 3) * 4;
  const float* ap0 = h + (size_t)(bm * BM + arow0) * DF;
  const float* ap1 = h + (size_t)(bm * BM + arow0 + 64) * DF;

  const int brow0 = tid >> 5;
  const int bc4 = (tid & 31) * 4;

  const int wid = tid >> 5;
  const int lane = tid & 31;
  const int wm = wid & 3;
  const int wn = wid >> 2;
  const int l16 = lane & 15;
  const int lhi = lane >> 4;

  v8f acc[2][4];
#pragma unroll
  for (int i = 0; i < 2; ++i)
#pragma unroll
    for (int j = 0; j < 4; ++j) acc[i][j] = (v8f)0.0f;

  for (int k0 = 0; k0 < DF; k0 += BK) {
    *(float4*)&lA[arow0 * LDA + ac4] = *(const float4*)(ap0 + k0 + ac4);
    *(float4*)&lA[(arow0 + 64) * LDA + ac4] = *(const float4*)(ap1 + k0 + ac4);
#pragma unroll
    for (int i = 0; i < 2; ++i) {
      int br = brow0 + i * 8;
      *(float4*)&lB[br * BN + bc4] = *(const float4*)&W2e[(size_t)(k0 + br) * DM + db + bc4];
    }
    __syncthreads();
#pragma unroll
    for (int kk = 0; kk < BK; kk += 4) {
      const int kin = kk + lhi * 2;
      v2f af[2], bf[4];
#pragma unroll
      for (int tm = 0; tm < 2; ++tm) {
        int r = wm * 32 + tm * 16 + l16;
        af[tm].x = lA[r * LDA + kin];
        af[tm].y = lA[r * LDA + kin + 1];
      }
#pragma unroll
      for (int tn = 0; tn < 4; ++tn) {
        int c = wn * 64 + tn * 16 + l16;
        bf[tn].x = lB[kin * BN + c];
        bf[tn].y = lB[(kin + 1) * BN + c];
      }
#pragma unroll
      for (int tm = 0; tm < 2; ++tm)
#pragma unroll
        for (int tn = 0; tn < 4; ++tn)
          acc[tm][tn] = __builtin_amdgcn_wmma_f32_16x16x4_f32(
              false, af[tm], false, bf[tn], (short)0, acc[tm][tn], false, false);
    }
    __syncthreads();
  }
  // scatter-add: each out element gets exactly 2 contributions -> deterministic
#pragma unroll
  for (int tm = 0; tm < 2; ++tm)
#pragma unroll
    for (int tn = 0; tn < 4; ++tn) {
      int rbase = bm * BM + wm * 32 + tm * 16 + lhi * 8;
      int cbase = db + wn * 64 + tn * 16 + l16;
#pragma unroll
      for (int v = 0; v < 8; ++v) {
        int r = rbase + v;
        int t = slot_token[r];
        if (t >= 0) {
          float w = slot_w[r];
          atomicAdd(&out[(size_t)t * DM + cbase], w * acc[tm][tn][v]);
        }
      }
    }
}

extern "C" void kernel_launch(void* const* d_in, const int* in_sizes, int n_in,
                              void* d_out, int out_size, void* d_ws, size_t ws_size,
                              hipStream_t stream) {
  const float* x  = (const float*)d_in[0];
  const float* Wr = (const float*)d_in[1];
  const float* W1 = (const float*)d_in[2];
  const float* W3 = (const float*)d_in[3];
  const float* W2 = (const float*)d_in[4];
  float* out = (float*)d_out;

  char* ws = (char*)d_ws;
  int*   counts       = (int*)(ws + 0);
  int*   cursors      = (int*)(ws + 64);
  int*   block_expert = (int*)(ws + 128);
  float* psum         = (float*)(ws + 512);
  int*   tk_i         = (int*)(ws + 1024);
  float* tk_w         = (float*)(ws + 1024 + NTOK * 2 * 4);
  int*   slot_token   = (int*)(ws + 40960);
  float* slot_w       = (float*)(ws + 40960 + MCAP * 4);
  float* h            = (float*)(ws + (1 << 17));   // 5120 x 4096 fp32 (~80 MB)

  hipMemsetAsync(counts, 0, 32, stream);
  hipMemsetAsync(psum, 0, 32, stream);
  hipMemsetAsync(slot_token, 0xFF, MCAP * 4, stream);          // all -1
  hipMemsetAsync(out, 0, (size_t)NTOK * DM * sizeof(float), stream);

  moe_router<<<NTOK, 256, 0, stream>>>(x, Wr, tk_i, tk_w, counts, psum);
  moe_prefix<<<1, 32, 0, stream>>>(counts, psum, cursors, block_expert,
                                   out + (size_t)NTOK * DM);
  moe_scatter<<<NTOK / 256, 256, 0, stream>>>(tk_i, tk_w, cursors, slot_token, slot_w);
  moe_gemm1<<<dim3(DF / BN, MBLK), 256, 0, stream>>>(x, W1, W3, slot_token,
                                                     block_expert, h);
  moe_gemm2<<<dim3(DM / BN, MBLK), 256, 0, stream>>>(h, W2, slot_token, slot_w,
                                                     block_expert, out);
}